// WaveletLayer_83167746720515
// MI455X (gfx1250) — compile-verified
//
#include <hip/hip_runtime.h>
#include <cstdint>

// Native 2-float vector so __builtin_nontemporal_store works on a b64 store.
typedef float f2 __attribute__((ext_vector_type(2)));

namespace {
// Fixed problem geometry from the reference: x[2,32,64,128,128] fp32.
constexpr unsigned Bn = 2, Cn = 32, Dn = 64, Hn = 128, Wn = 128;
constexpr unsigned HW   = Hn * Wn;            // 16384
constexpr unsigned DHW  = Dn * HW;            // 1,048,576 elements per (b,c) slab
constexpr unsigned OD = Dn / 2, OH = Hn / 2, OW = Wn / 2;  // 32, 64, 64
constexpr unsigned OCHW   = OD * OH * OW;     // 131072 elements per output channel
constexpr unsigned OBATCH = 8u * Cn * OCHW;   // 33,554,432 elements per batch
constexpr unsigned SBSTR  = Cn * OCHW;        // 4,194,304: stride between subbands

constexpr unsigned NU    = 4;                                    // units per thread
constexpr unsigned UNITS = Bn * Cn * OD * OH * (OW / 2);         // 4,194,304
constexpr unsigned TT    = UNITS / NU;                           // 1,048,576 threads
constexpr unsigned TPB   = 256;                                  // 8 wave32s
constexpr unsigned NBLK  = TT / TPB;                             // 4096 blocks
}

// 8-point separable Haar butterfly: inputs x[z][y][w], outputs o[s],
// s = 4*z_hi + 2*y_hi + w_hi  (matches reference concat order aaa..ddd).
__device__ __forceinline__ void haar8(float x000, float x001, float x010, float x011,
                                      float x100, float x101, float x110, float x111,
                                      float o[8]) {
  const float K = 0.35355339059327373f;  // (1/sqrt(2))^3
  // W stage: Wwh_zy
  float W000 = x000 + x001, W100 = x000 - x001;  // z0 y0
  float W001 = x010 + x011, W101 = x010 - x011;  // z0 y1
  float W010 = x100 + x101, W110 = x100 - x101;  // z1 y0
  float W011 = x110 + x111, W111 = x110 - x111;  // z1 y1
  // H stage: Hwh_yh_z
  float H000 = W000 + W001, H010 = W000 - W001;  // wh=0, z0
  float H001 = W010 + W011, H011 = W010 - W011;  // wh=0, z1
  float H100 = W100 + W101, H110 = W100 - W101;  // wh=1, z0
  float H101 = W110 + W111, H111 = W110 - W111;  // wh=1, z1
  // D stage
  o[0] = (H000 + H001) * K;  // aaa
  o[1] = (H100 + H101) * K;  // aad
  o[2] = (H010 + H011) * K;  // ada
  o[3] = (H110 + H111) * K;  // add
  o[4] = (H000 - H001) * K;  // daa
  o[5] = (H100 - H101) * K;  // dad
  o[6] = (H010 - H011) * K;  // dda
  o[7] = (H110 - H111) * K;  // ddd
}

__global__ __launch_bounds__(TPB) void haar3d_dwt_async_kernel(
    const float* __restrict__ x, float* __restrict__ out) {
  // Per-thread double-buffered staging: [buf][row][tid] so ds_load_b128 lanes
  // stride 16B -> conflict-minimal. 2*4*256*16B = 32 KiB.
  __shared__ float4 sh[2][4][TPB];

  const unsigned tid = threadIdx.x;
  const unsigned g   = blockIdx.x * TPB + tid;

  // Unit u encodes (bc, dz, dy, p): p = w-pair index (2 output blocks wide).
  auto issue = [&](unsigned u, int buf) {
    const unsigned p  = u & 31u;
    const unsigned dy = (u >> 5) & 63u;
    const unsigned dz = (u >> 11) & 31u;
    const unsigned bc = u >> 16;  // 0..63
    const float* base = x + (bc * DHW + dz * (2u * HW) + dy * (2u * Wn) + p * 4u);

    // LDS byte offsets (low 32 bits of the flat LDS address == wave LDS offset).
    const uint32_t l0 = (uint32_t)(uintptr_t)&sh[buf][0][tid];
    const uint32_t l1 = (uint32_t)(uintptr_t)&sh[buf][1][tid];
    const uint32_t l2 = (uint32_t)(uintptr_t)&sh[buf][2][tid];
    const uint32_t l3 = (uint32_t)(uintptr_t)&sh[buf][3][tid];
    const unsigned long long a0 = (unsigned long long)(uintptr_t)(base);            // z0 y0
    const unsigned long long a1 = (unsigned long long)(uintptr_t)(base + Wn);       // z0 y1
    const unsigned long long a2 = (unsigned long long)(uintptr_t)(base + HW);       // z1 y0
    const unsigned long long a3 = (unsigned long long)(uintptr_t)(base + HW + Wn);  // z1 y1

    // CDNA5 async global->LDS copies (tracked by ASYNCcnt, no dest VGPRs).
    asm volatile("global_load_async_to_lds_b128 %0, %1, off" ::"v"(l0), "v"(a0) : "memory");
    asm volatile("global_load_async_to_lds_b128 %0, %1, off" ::"v"(l1), "v"(a1) : "memory");
    asm volatile("global_load_async_to_lds_b128 %0, %1, off" ::"v"(l2), "v"(a2) : "memory");
    asm volatile("global_load_async_to_lds_b128 %0, %1, off" ::"v"(l3), "v"(a3) : "memory");
  };

  auto compute = [&](unsigned u, int buf) {
    const unsigned p  = u & 31u;
    const unsigned dy = (u >> 5) & 63u;
    const unsigned dz = (u >> 11) & 31u;
    const unsigned bc = u >> 16;
    const unsigned b  = bc >> 5;
    const unsigned c  = bc & 31u;

    const float4 r00 = sh[buf][0][tid];  // z0 y0, w = 4p..4p+3
    const float4 r01 = sh[buf][1][tid];  // z0 y1
    const float4 r10 = sh[buf][2][tid];  // z1 y0
    const float4 r11 = sh[buf][3][tid];  // z1 y1

    float oA[8], oB[8];
    haar8(r00.x, r00.y, r01.x, r01.y, r10.x, r10.y, r11.x, r11.y, oA);  // block dx=2p
    haar8(r00.z, r00.w, r01.z, r01.w, r10.z, r10.w, r11.z, r11.w, oB);  // block dx=2p+1

    const unsigned obase = b * OBATCH + c * OCHW + dz * (OH * OW) + dy * OW + p * 2u;
#pragma unroll
    for (unsigned s = 0; s < 8; ++s) {
      f2 v;
      v.x = oA[s];
      v.y = oB[s];
      // Write-once streaming output: NT store, don't thrash the 192 MB L2.
      __builtin_nontemporal_store(v, reinterpret_cast<f2*>(out + obase + s * SBSTR));
    }
  };

  // Software pipeline over NU units, double-buffered in LDS via ASYNCcnt.
  issue(g, 0);
#pragma unroll
  for (unsigned i = 0; i < NU; ++i) {
    if (i + 1 < NU) {
      issue(g + (i + 1) * TT, (int)((i + 1) & 1u));
      // 8 async ops outstanding; <=4 left means current buffer's 4 completed
      // (async loads complete in order).
      asm volatile("s_wait_asynccnt 4" ::: "memory");
    } else {
      asm volatile("s_wait_asynccnt 0" ::: "memory");
    }
    compute(g + i * TT, (int)(i & 1u));
  }
}

extern "C" void kernel_launch(void* const* d_in, const int* in_sizes, int n_in,
                              void* d_out, int out_size, void* d_ws, size_t ws_size,
                              hipStream_t stream) {
  (void)in_sizes; (void)n_in; (void)d_ws; (void)ws_size; (void)out_size;
  const float* x = (const float*)d_in[0];
  float* out = (float*)d_out;
  hipLaunchKernelGGL(haar3d_dwt_async_kernel, dim3(NBLK), dim3(TPB), 0, stream, x, out);
}